// LearnableWaveletDecomposition_88218628259890
// MI455X (gfx1250) — compile-verified
//
#include <hip/hip_runtime.h>
#include <hip/hip_bf16.h>
#include <math.h>

// ---------------------------------------------------------------------------
// Problem constants (from the reference)
// ---------------------------------------------------------------------------
#define D_MODEL 1024
#define LEVELS  3
#define WL      8
#define BATCH   16
#define LSEQ    4096

// Decomposition lengths: n_out = n/2 + 1 per level
#define N0 2049   // hi level 0
#define N1 1025   // hi level 1
#define N2 513    // hi level 2
#define N3 513    // lo level 2 (approx)

typedef __attribute__((ext_vector_type(2))) float v2f;
typedef __attribute__((ext_vector_type(8))) float v8f;

// ---------------------------------------------------------------------------
// Helpers
// ---------------------------------------------------------------------------
__device__ __forceinline__ void softmax4(const float* __restrict__ cw, float w[4]) {
    float c0 = cw[0], c1 = cw[1], c2 = cw[2], c3 = cw[3];
    float m = fmaxf(fmaxf(c0, c1), fmaxf(c2, c3));
    float e0 = __expf(c0 - m), e1 = __expf(c1 - m), e2 = __expf(c2 - m), e3 = __expf(c3 - m);
    float inv = 1.0f / (e0 + e1 + e2 + e3);
    w[0] = e0 * inv; w[1] = e1 * inv; w[2] = e2 * inv; w[3] = e3 * inv;
}

// One wavelet level: reflect-pad by 4, correlate with 8-tap low/high filters,
// keep even output positions (j = 0,2,...,2*(n/2)). Outputs n/2+1 values.
__device__ __forceinline__ void wav_level(const float* __restrict__ in, int n,
                                          float* __restrict__ outLo,
                                          float* __restrict__ outHi,
                                          const float lf[WL], const float hf[WL],
                                          int tid, int nthreads) {
    int n2 = n / 2 + 1;
    for (int jj = tid; jj < n2; jj += nthreads) {
        int j = 2 * jj;
        float accL = 0.0f, accH = 0.0f;
#pragma unroll
        for (int t = 0; t < WL; ++t) {
            int m = j + t - (WL / 2);
            if (m < 0) m = -m;                       // reflect (no edge repeat)
            else if (m >= n) m = 2 * (n - 1) - m;
            float v = in[m];
            accL = fmaf(v, lf[t], accL);
            accH = fmaf(v, hf[t], accH);
        }
        outLo[jj] = accL;
        outHi[jj] = accH;
    }
}

// Linear resize sample of a length-n signal (in LDS) at output position l of LSEQ.
__device__ __forceinline__ float resize_at(const float* __restrict__ d, int n, int l) {
    float s = ((float)l + 0.5f) * ((float)n / (float)LSEQ) - 0.5f;
    s = fminf(fmaxf(s, 0.0f), (float)(n - 1));
    int i0 = (int)s;                  // s >= 0 -> trunc == floor
    int i1 = min(i0 + 1, n - 1);
    float fr = s - (float)i0;
    return d[i0] * (1.0f - fr) + d[i1] * fr;
}

// ---------------------------------------------------------------------------
// Kernel 1: per-batch wavelet decomposition + resize, entirely in LDS.
// Emits c[b][l][4] = softmax(comb_w)[i] * resize_i(b, l)   (pre-scaled A matrix)
// ---------------------------------------------------------------------------
__global__ __launch_bounds__(256)
void wavelet_coeff_kernel(const float* __restrict__ x,       // [B][L]
                          const float* __restrict__ lowf,    // [LEVELS][WL]
                          const float* __restrict__ highf,   // [LEVELS][WL]
                          const float* __restrict__ comb_w,  // [4]
                          float* __restrict__ c_out) {       // [B][L][4]
    __shared__ float sig[LSEQ];     // level-0 input; reused as level-2 input
    __shared__ float lo1[N0];       // level-0 approx -> level-1 input
    __shared__ float dec0[N0];
    __shared__ float dec1[N1];
    __shared__ float dec2[N2];
    __shared__ float dec3[N3];

    const int b   = blockIdx.x;
    const int tid = threadIdx.x;
    const int nt  = blockDim.x;

    // load signal
    for (int i = tid; i < LSEQ; i += nt) sig[i] = x[(size_t)b * LSEQ + i];

    float lf[WL], hf[WL];

    // level 0: sig(4096) -> lo1(2049), dec0(2049)
#pragma unroll
    for (int t = 0; t < WL; ++t) { lf[t] = lowf[0 * WL + t]; hf[t] = highf[0 * WL + t]; }
    __syncthreads();
    wav_level(sig, LSEQ, lo1, dec0, lf, hf, tid, nt);
    __syncthreads();

    // level 1: lo1(2049) -> sig[0..1024](1025), dec1(1025)   (sig is dead now)
#pragma unroll
    for (int t = 0; t < WL; ++t) { lf[t] = lowf[1 * WL + t]; hf[t] = highf[1 * WL + t]; }
    wav_level(lo1, N0, sig, dec1, lf, hf, tid, nt);
    __syncthreads();

    // level 2: sig(1025) -> dec3(513) approx, dec2(513) detail
#pragma unroll
    for (int t = 0; t < WL; ++t) { lf[t] = lowf[2 * WL + t]; hf[t] = highf[2 * WL + t]; }
    wav_level(sig, N1, dec3, dec2, lf, hf, tid, nt);
    __syncthreads();

    // resize all four bands to LSEQ, scale by softmax weights, store float4
    float w[4];
    softmax4(comb_w, w);
    for (int l = tid; l < LSEQ; l += nt) {
        float4 cv;
        cv.x = w[0] * resize_at(dec0, N0, l);
        cv.y = w[1] * resize_at(dec1, N1, l);
        cv.z = w[2] * resize_at(dec2, N2, l);
        cv.w = w[3] * resize_at(dec3, N3, l);
        *reinterpret_cast<float4*>(c_out + (((size_t)b * LSEQ + l) << 2)) = cv;
    }
}

// ---------------------------------------------------------------------------
// Kernel 2: rank-4 projection via V_WMMA_F32_16X16X4_F32.
//   out[b, l, d] = sum_k c[b][l][k] * proj_w[k][d] + sum_k w[k]*proj_b[k][d]
// One wave = one 16(L) x 64(D) strip: A reused across 4 WMMAs.
// Waves: B * (L/16) * (D/64) = 16 * 256 * 16 = 65536 -> 8192 blocks x 256 thr.
// ---------------------------------------------------------------------------
__global__ __launch_bounds__(256)
void wmma_proj_kernel(const float* __restrict__ c,       // [B][L][4]  (pre-scaled)
                      const float* __restrict__ proj_w,  // [4][D_MODEL]
                      const float* __restrict__ proj_b,  // [4][D_MODEL]
                      const float* __restrict__ comb_w,  // [4]
                      float* __restrict__ out) {         // [B][L][D_MODEL]
    const int gid   = blockIdx.x * blockDim.x + threadIdx.x;
    const int lane  = threadIdx.x & 31;
    const int wave  = gid >> 5;

    const int dg = wave & 15;          // which group of 64 d-columns
    const int lt = (wave >> 4) & 255;  // which 16-row L tile
    const int b  = wave >> 12;         // batch
    const int l0 = lt << 4;

    float w[4];
    softmax4(comb_w, w);

    const int m     = lane & 15;   // row within tile (A) / column (B,C,D)
    const int khalf = lane >> 4;   // 0: K=0,1   1: K=2,3

    // A-matrix 16x4 f32 layout: lanes 0-15 hold {K0,K1}, lanes 16-31 hold {K2,K3}
    const float* cptr = c + ((((size_t)b * LSEQ) + l0 + m) << 2) + (khalf << 1);
    v2f a;
    a.x = cptr[0];
    a.y = cptr[1];

    const size_t row_base = ((size_t)b * LSEQ + l0 + (khalf << 3)) * D_MODEL;

#pragma unroll
    for (int dt = 0; dt < 4; ++dt) {
        const int col = (dg << 6) + (dt << 4) + m;

        // B-matrix 4x16 f32: VGPR j holds row K = khalf*2 + j, column = lane&15
        v2f bm;
        bm.x = proj_w[(size_t)((khalf << 1) + 0) * D_MODEL + col];
        bm.y = proj_w[(size_t)((khalf << 1) + 1) * D_MODEL + col];

        // bias depends only on N (= lane&15) -> replicate across all 8 C rows
        float bias = w[0] * proj_b[col]
                   + w[1] * proj_b[1 * D_MODEL + col]
                   + w[2] * proj_b[2 * D_MODEL + col]
                   + w[3] * proj_b[3 * D_MODEL + col];
        v8f cacc;
#pragma unroll
        for (int r = 0; r < 8; ++r) cacc[r] = bias;

        v8f d = __builtin_amdgcn_wmma_f32_16x16x4_f32(
            /*neg_a=*/false, a, /*neg_b=*/false, bm,
            /*c_mod=*/(short)0, cacc, /*reuse_a=*/false, /*reuse_b=*/false);

        // D layout: VGPR r -> row M = r + khalf*8, col N = lane&15
        float* op = out + row_base + col;
#pragma unroll
        for (int r = 0; r < 8; ++r)
            __builtin_nontemporal_store(d[r], op + (size_t)r * D_MODEL);
    }
}

// ---------------------------------------------------------------------------
// Launch
// ---------------------------------------------------------------------------
extern "C" void kernel_launch(void* const* d_in, const int* in_sizes, int n_in,
                              void* d_out, int out_size, void* d_ws, size_t ws_size,
                              hipStream_t stream) {
    const float* x      = (const float*)d_in[0];   // [16,4096,1]
    const float* lowf   = (const float*)d_in[1];   // [3,8]
    const float* highf  = (const float*)d_in[2];   // [3,8]
    const float* proj_w = (const float*)d_in[3];   // [4,1024]
    const float* proj_b = (const float*)d_in[4];   // [4,1024]
    const float* comb_w = (const float*)d_in[5];   // [4]
    float* out = (float*)d_out;                    // [16,4096,1024]

    float* c_ws = (float*)d_ws;                    // [16][4096][4] = 1 MB

    wavelet_coeff_kernel<<<BATCH, 256, 0, stream>>>(x, lowf, highf, comb_w, c_ws);

    const int total_waves = BATCH * (LSEQ / 16) * (D_MODEL / 64);   // 65536
    const int blocks = total_waves / 8;                             // 8 waves/block
    wmma_proj_kernel<<<blocks, 256, 0, stream>>>(c_ws, proj_w, proj_b, comb_w, out);
}